// HLDA_DualSpatial_46359876993195
// MI455X (gfx1250) — compile-verified
//
#include <hip/hip_runtime.h>
#include <math.h>

typedef __attribute__((ext_vector_type(2))) float v2f;
typedef __attribute__((ext_vector_type(8))) float v8f;

#define HWSZ (512 * 512)        // pixels per (b, plane)

__device__ __forceinline__ float sigmoidf_dev(float v) {
    return 1.0f / (1.0f + expf(-v));
}

// ---------------------------------------------------------------------------
// Pass 1: streaming channel reduction. Reads 256MB of x once (the only
// bandwidth-significant pass), produces l2map [B,512,512] and
// cat = [B,2,512,512] (max, mean). float4 per thread, coalesced; software
// prefetch covers the 1MB channel stride.
// ---------------------------------------------------------------------------
__global__ __launch_bounds__(256) void reduce_kernel(const float* __restrict__ x,
                                                     float* __restrict__ l2map,
                                                     float* __restrict__ cat) {
    int tid = blockIdx.x * blockDim.x + threadIdx.x;   // 262144 threads total
    int b = tid >> 16;            // HWSZ/4 = 65536 float4 per batch-plane
    int p = tid & 65535;
    const float4* px = (const float4*)x + (size_t)b * (64 * 65536) + p;

    float4 ssq = make_float4(0.f, 0.f, 0.f, 0.f);
    float4 sm  = make_float4(0.f, 0.f, 0.f, 0.f);
    float4 mx  = make_float4(-3.4e38f, -3.4e38f, -3.4e38f, -3.4e38f);

#pragma unroll 8
    for (int c = 0; c < 64; ++c) {
        if (c < 56) __builtin_prefetch(&px[(c + 8) * 65536], 0, 0);
        float4 v = px[c * 65536];
        ssq.x = fmaf(v.x, v.x, ssq.x); ssq.y = fmaf(v.y, v.y, ssq.y);
        ssq.z = fmaf(v.z, v.z, ssq.z); ssq.w = fmaf(v.w, v.w, ssq.w);
        sm.x += v.x; sm.y += v.y; sm.z += v.z; sm.w += v.w;
        mx.x = fmaxf(mx.x, v.x); mx.y = fmaxf(mx.y, v.y);
        mx.z = fmaxf(mx.z, v.z); mx.w = fmaxf(mx.w, v.w);
    }
    float4 l2 = make_float4(sqrtf(ssq.x), sqrtf(ssq.y), sqrtf(ssq.z), sqrtf(ssq.w));
    const float inv = 1.0f / 64.0f;
    float4 mean = make_float4(sm.x * inv, sm.y * inv, sm.z * inv, sm.w * inv);

    ((float4*)l2map)[(size_t)b * 65536 + p] = l2;
    ((float4*)cat)[((size_t)b * 2 + 0) * 65536 + p] = mx;
    ((float4*)cat)[((size_t)b * 2 + 1) * 65536 + p] = mean;
}

// ---------------------------------------------------------------------------
// Pass 2: local attention conv as GEMM with V_WMMA_F32_16X16X4_F32.
// A = 16 pixels x K (K = 4 in-tiles * 9 taps = 36), B = K x 16 out-channels
// (4 valid). 9 WMMA steps per 16-pixel tile. One wave per block; the wave
// handles 4 tiles along ws so the B matrix (weights) is built once.
//
// Border handling is branchless: clamp coordinates into the tile (address
// always valid, all lanes load -> no EXEC churn before the WMMA), then
// select 0.0 for out-of-bounds taps (zero padding).
// ---------------------------------------------------------------------------
__device__ __forceinline__ float fetchA_local(const float* __restrict__ L,
                                              int hs, int xc, int k) {
    int ci = k / 9;
    int kk = k - ci * 9;
    int y  = hs + kk / 3 - 1;
    int xx = xc + (kk % 3) - 1;
    bool ok = ((unsigned)y < 256u) & ((unsigned)xx < 256u);
    int yc  = y & 255;             // in-bounds value unchanged; OOB -> safe addr
    int xxc = xx & 255;
    float v = L[(((ci >> 1) << 8) + yc) * 512 + ((ci & 1) << 8) + xxc];
    return ok ? v : 0.f;
}

__global__ __launch_bounds__(32) void local_conv_kernel(const float* __restrict__ l2map,
                                                        const float* __restrict__ lw,
                                                        const float* __restrict__ lb,
                                                        float* __restrict__ out) {
    int lane = threadIdx.x & 31;
    int hi   = lane >> 4;
    int mn   = lane & 15;          // M row for A, N column for B/D
    int koff = hi ? 2 : 0;

    int bid = blockIdx.x;          // 4096 blocks = B(4) * hs(256) * quarter(4)
    int q  = bid & 3;
    int hs = (bid >> 2) & 255;
    int b  = bid >> 10;

    // Build B (weights) once per wave: lane = out-channel column.
    // Branchless: clamp channel to [0,3], select 0 for n >= 4.
    int nc = mn & 3;
    bool nok = (mn < 4);
    v2f Bv[9];
#pragma unroll
    for (int s = 0; s < 9; ++s) {
        int k0 = 4 * s + koff;
        int k = k0, ci = k / 9, kk = k - ci * 9;
        float b0 = lw[((nc * 4 + ci) * 3 + kk / 3) * 3 + (kk % 3)];
        k = k0 + 1; ci = k / 9; kk = k - ci * 9;
        float b1 = lw[((nc * 4 + ci) * 3 + kk / 3) * 3 + (kk % 3)];
        Bv[s].x = nok ? b0 : 0.f;
        Bv[s].y = nok ? b1 : 0.f;
    }
    float bias = nok ? lb[nc] : 0.f;
    const float* L = l2map + (size_t)b * HWSZ;

    for (int t = 0; t < 4; ++t) {
        int ws0 = q * 64 + t * 16;
        v8f acc = {0.f, 0.f, 0.f, 0.f, 0.f, 0.f, 0.f, 0.f};
#pragma unroll
        for (int s = 0; s < 9; ++s) {
            int k0 = 4 * s + koff;
            v2f a;
            a.x = fetchA_local(L, hs, ws0 + mn, k0);
            a.y = fetchA_local(L, hs, ws0 + mn, k0 + 1);
            acc = __builtin_amdgcn_wmma_f32_16x16x4_f32(
                false, a, false, Bv[s], (short)0, acc, false, false);
        }
        if (mn < 4) {
            // reshape remap: (c,hs,ws) -> h = c*128 + hs/2, w = (hs&1)*256 + ws
            int h = mn * 128 + (hs >> 1);
            float* o = out + (size_t)b * HWSZ + (size_t)h * 512 + ((hs & 1) << 8) + ws0;
#pragma unroll
            for (int r = 0; r < 8; ++r) {
                int mm = r + hi * 8;
                o[mm] = sigmoidf_dev(acc[r] + bias);
            }
        }
    }
}

// ---------------------------------------------------------------------------
// Pass 3: global attention conv as GEMM (K = 2*9 = 18, padded to 20 -> 5 WMMA
// steps; only N=0 valid). Adds sigmoid(relu(conv+bias)) into d_out.
// Same branchless clamp+select border handling; padded K-steps clamp ci to 1
// so no address ever leaves the cat buffer.
// ---------------------------------------------------------------------------
__device__ __forceinline__ float fetchA_global(const float* __restrict__ Cc,
                                               int h, int xc, int k) {
    int ci = k / 9;
    int kk = k - ci * 9;
    int y  = h + kk / 3 - 1;
    int xx = xc + (kk % 3) - 1;
    bool ok = (k < 18) & ((unsigned)y < 512u) & ((unsigned)xx < 512u);
    int cic = (ci < 2) ? ci : 1;   // keep padded K-steps inside the buffer
    int yc  = y & 511;
    int xxc = xx & 511;
    float v = Cc[(size_t)cic * HWSZ + yc * 512 + xxc];
    return ok ? v : 0.f;
}

__global__ __launch_bounds__(32) void global_conv_kernel(const float* __restrict__ cat,
                                                         const float* __restrict__ gw,
                                                         const float* __restrict__ gb,
                                                         float* __restrict__ out) {
    int lane = threadIdx.x & 31;
    int hi   = lane >> 4;
    int mn   = lane & 15;
    int koff = hi ? 2 : 0;

    int bid = blockIdx.x;          // 8192 blocks = B(4) * h(512) * quarter(4)
    int q = bid & 3;
    int h = (bid >> 2) & 511;
    int b = bid >> 11;

    v2f Bv[5];
#pragma unroll
    for (int s = 0; s < 5; ++s) {
        int k0 = 4 * s + koff;
        float b0 = gw[(k0 < 18) ? k0 : 0];
        float b1 = gw[(k0 + 1 < 18) ? (k0 + 1) : 0];
        bool ok0 = (mn == 0) & (k0 < 18);
        bool ok1 = (mn == 0) & (k0 + 1 < 18);
        Bv[s].x = ok0 ? b0 : 0.f;
        Bv[s].y = ok1 ? b1 : 0.f;
    }
    float bias0 = gb[0];
    const float* Cc = cat + (size_t)b * 2 * HWSZ;

    for (int t = 0; t < 8; ++t) {
        int w0 = q * 128 + t * 16;
        v8f acc = {0.f, 0.f, 0.f, 0.f, 0.f, 0.f, 0.f, 0.f};
#pragma unroll
        for (int s = 0; s < 5; ++s) {
            int k0 = 4 * s + koff;
            v2f a;
            a.x = fetchA_global(Cc, h, w0 + mn, k0);
            a.y = fetchA_global(Cc, h, w0 + mn, k0 + 1);
            acc = __builtin_amdgcn_wmma_f32_16x16x4_f32(
                false, a, false, Bv[s], (short)0, acc, false, false);
        }
        if (mn == 0) {
            float* o = out + (size_t)b * HWSZ + (size_t)h * 512 + w0;
#pragma unroll
            for (int r = 0; r < 8; ++r) {
                int mm = r + hi * 8;
                float v = fmaxf(acc[r] + bias0, 0.f);
                o[mm] += sigmoidf_dev(v);
            }
        }
    }
}

// ---------------------------------------------------------------------------
extern "C" void kernel_launch(void* const* d_in, const int* in_sizes, int n_in,
                              void* d_out, int out_size, void* d_ws, size_t ws_size,
                              hipStream_t stream) {
    const float* x  = (const float*)d_in[0];   // [4,64,512,512]
    const float* lw = (const float*)d_in[1];   // [4,4,3,3]
    const float* lb = (const float*)d_in[2];   // [4]
    const float* gw = (const float*)d_in[3];   // [1,2,3,3]
    const float* gb = (const float*)d_in[4];   // [1]
    float* out = (float*)d_out;                // [4,1,512,512]

    float* l2map = (float*)d_ws;               // 4*HWSZ floats (4 MB)
    float* cat   = l2map + (size_t)4 * HWSZ;   // 8*HWSZ floats (8 MB)

    reduce_kernel<<<1024, 256, 0, stream>>>(x, l2map, cat);
    local_conv_kernel<<<4096, 32, 0, stream>>>(l2map, lw, lb, out);
    global_conv_kernel<<<8192, 32, 0, stream>>>(cat, gw, gb, out);
}